// VolumeRenderingGeneralModule_10033043603888
// MI455X (gfx1250) — compile-verified
//
#include <hip/hip_runtime.h>

// Volume rendering composite for MI455X (gfx1250, wave32).
// Bandwidth-bound: ~240MB of unavoidable traffic @ 23.3 TB/s -> ~10us floor;
// arithmetic intensity ~0.1 FLOP/byte, so WMMA is irrelevant — the CDNA5
// feature that matters is the ASYNCcnt-tracked global->LDS DMA path.
//   kernel 1 (render): one wave32 per ray; 32-sample chunks staged into a
//     per-wave LDS double buffer with global_load_async_to_lds_b32; inclusive
//     wave-shuffle scan gives the segmented transmittance prefix; shuffle
//     reduction produces the per-ray outputs.
//   kernel 2 (row_ptr): CSR boundaries from sorted int64 ray_id via binary
//     search (handles empty rays: start==end -> rgb/depth=0, bg=exp(0)=1).

#define WPB   8            // waves per block
#define LANES 32
#define CHUNK_FLOATS 192   // 32 sigma + 32 z + 32 dt + 96 rgb

__device__ __forceinline__ void async_ld_b32(unsigned lds_byte_addr, const float* g) {
    unsigned long long ga = (unsigned long long)(const void*)g;
    // CDNA5 async global->LDS DMA (ASYNCcnt). VDST VGPR = LDS byte address.
    asm volatile("global_load_async_to_lds_b32 %0, %1, off"
                 :: "v"(lds_byte_addr), "v"(ga) : "memory");
}

__device__ __forceinline__ void issue_chunk(int base, int e, int lane, unsigned lds,
                                            const float* __restrict__ sig,
                                            const float* __restrict__ zs,
                                            const float* __restrict__ dts,
                                            const float* __restrict__ rgb) {
    int last = e - 1;
    int i = base + lane; if (i > last) i = last;           // clamp: stay in segment
    async_ld_b32(lds + 4u * (unsigned)lane,        sig + i);
    async_ld_b32(lds + 4u * (unsigned)(32 + lane), zs  + i);
    async_ld_b32(lds + 4u * (unsigned)(64 + lane), dts + i);
    int dlast = 3 * e - 1;
#pragma unroll
    for (int j = 0; j < 3; ++j) {                           // 96 contiguous rgb dwords
        int d = 3 * base + 32 * j + lane; if (d > dlast) d = dlast;
        async_ld_b32(lds + 4u * (unsigned)(96 + 32 * j + lane), rgb + d);
    }
}

__global__ void __launch_bounds__(WPB * LANES)
render_kernel(const float* __restrict__ rgb, const float* __restrict__ sig,
              const float* __restrict__ zs,  const float* __restrict__ dts,
              const float* __restrict__ t_exit, const int* __restrict__ use_flag,
              const int* __restrict__ row_ptr,
              float* __restrict__ o_rgb, float* __restrict__ o_depth,
              float* __restrict__ o_bg,  float* __restrict__ o_w, int n_rays) {
    __shared__ float smem[WPB * 2 * CHUNK_FLOATS];
    const int lane = threadIdx.x & 31;
    const int wid  = threadIdx.x >> 5;
    const int q    = blockIdx.x * WPB + wid;
    if (q >= n_rays) return;

    const int s = row_ptr[q];
    const int e = row_ptr[q + 1];
    const int use_exit = use_flag[0];
    const float tq = t_exit[q];

    float* buf0 = &smem[(wid * 2 + 0) * CHUNK_FLOATS];
    float* buf1 = &smem[(wid * 2 + 1) * CHUNK_FLOATS];
    const unsigned a0 = (unsigned)(unsigned long long)(const void*)buf0;
    const unsigned a1 = (unsigned)(unsigned long long)(const void*)buf1;

    float carry = 0.f, aR = 0.f, aG = 0.f, aB = 0.f, aD = 0.f;

    if (s < e) {
        issue_chunk(s, e, lane, a0, sig, zs, dts, rgb);     // prologue: chunk 0
        int cur = 0;
#pragma unroll 2
        for (int base = s; base < e; base += LANES) {
            const float* bc = cur ? buf1 : buf0;
            int nbase = base + LANES;
            if (nbase < e) {
                issue_chunk(nbase, e, lane, cur ? a0 : a1, sig, zs, dts, rgb);
                asm volatile("s_wait_asynccnt 6" ::: "memory");  // current chunk landed
            } else {
                asm volatile("s_wait_asynccnt 0" ::: "memory");
            }
            int  i   = base + lane;
            bool act = i < e;
            float sg  = bc[lane];
            float zv  = bc[32 + lane];
            float dtv = bc[64 + lane];
            float r   = bc[96 + 3 * lane + 0];
            float g   = bc[96 + 3 * lane + 1];
            float b   = bc[96 + 3 * lane + 2];
            if (use_exit && i == e - 1) dtv = tq - zv;       // last-sample dt override
            float sigdt = act ? sg * dtv : 0.f;

            // wave32 inclusive scan of sigdt
            float x = sigdt;
#pragma unroll
            for (int off = 1; off < 32; off <<= 1) {
                float y = __shfl_up(x, off, 32);
                if (lane >= off) x += y;
            }
            float excl  = carry + (x - sigdt);               // segmented exclusive prefix
            float T     = __expf(-excl);
            float alpha = 1.f - __expf(-sigdt);
            float w     = alpha * T;
            if (act) {
                o_w[i] = w;
                aR += w * r; aG += w * g; aB += w * b; aD += w * zv;
            }
            carry += __shfl(x, 31, 32);                      // segment running total
            cur ^= 1;
        }
    }

    // wave reduction of per-lane accumulators
#pragma unroll
    for (int off = 16; off > 0; off >>= 1) {
        aR += __shfl_down(aR, off, 32);
        aG += __shfl_down(aG, off, 32);
        aB += __shfl_down(aB, off, 32);
        aD += __shfl_down(aD, off, 32);
    }
    if (lane == 0) {
        o_rgb[3 * q + 0] = aR;
        o_rgb[3 * q + 1] = aG;
        o_rgb[3 * q + 2] = aB;
        o_depth[q] = aD;
        o_bg[q] = __expf(-carry);                            // empty ray -> exp(0)=1
    }
}

__global__ void row_ptr_kernel(const long long* __restrict__ ray_id, int n,
                               int* __restrict__ row_ptr, int n_rays) {
    int q = blockIdx.x * blockDim.x + threadIdx.x;
    if (q > n_rays) return;
    if (q == n_rays) { row_ptr[q] = n; return; }
    // lower_bound: first i with ray_id[i] >= q
    int lo = 0, hi = n;
    while (lo < hi) {
        int mid = (lo + hi) >> 1;
        if (ray_id[mid] < (long long)q) lo = mid + 1; else hi = mid;
    }
    row_ptr[q] = lo;
}

extern "C" void kernel_launch(void* const* d_in, const int* in_sizes, int n_in,
                              void* d_out, int out_size, void* d_ws, size_t ws_size,
                              hipStream_t stream) {
    (void)n_in; (void)out_size; (void)ws_size;
    const float*     rgb      = (const float*)d_in[0];
    const float*     rad      = (const float*)d_in[1];      // (N,1) -> sigma
    const float*     zs       = (const float*)d_in[2];
    const float*     dts      = (const float*)d_in[3];
    const long long* ray_id   = (const long long*)d_in[4];  // int64 per reference
    const float*     t_exit   = (const float*)d_in[5];
    const int*       use_flag = (const int*)d_in[6];

    const int n      = in_sizes[2];   // N_SAMPLES
    const int n_rays = in_sizes[5];   // N_RAYS

    int* row_ptr = (int*)d_ws;        // (n_rays+1) ints of scratch

    float* out     = (float*)d_out;
    float* o_rgb   = out;                              // (n_rays,3)
    float* o_depth = out + 3 * (size_t)n_rays;         // (n_rays,1)
    float* o_bg    = o_depth + n_rays;                 // (n_rays,1)
    float* o_w     = o_bg + n_rays;                    // (N,1)

    const int t1 = 256;
    const int g1 = (n_rays + 1 + t1 - 1) / t1;
    row_ptr_kernel<<<g1, t1, 0, stream>>>(ray_id, n, row_ptr, n_rays);

    const int g2 = (n_rays + WPB - 1) / WPB;
    render_kernel<<<g2, WPB * LANES, 0, stream>>>(rgb, rad, zs, dts, t_exit, use_flag,
                                                  row_ptr, o_rgb, o_depth, o_bg, o_w,
                                                  n_rays);
}